// SparseConv3dTestTorch_50483045597963
// MI455X (gfx1250) — compile-verified
//
#include <hip/hip_runtime.h>

#define KS     3
#define PADV   1
#define OUTD   64
#define CIN    32
#define COUT   64

typedef __attribute__((ext_vector_type(2))) float v2f;
typedef __attribute__((ext_vector_type(8))) float v8f;

// ---------------------------------------------------------------------------
// Zero the 128 MB dense output (required every call: we accumulate atomically).
// ---------------------------------------------------------------------------
__global__ void spconv_zero_kernel(float4* __restrict__ out, int n4) {
    int i = blockIdx.x * blockDim.x + threadIdx.x;
    const int stride = gridDim.x * blockDim.x;
    const float4 z = make_float4(0.f, 0.f, 0.f, 0.f);
    for (; i < n4; i += stride) out[i] = z;
}

// ---------------------------------------------------------------------------
// Gather -> WMMA(f32 16x16x4) -> scatter-atomic sparse conv.
// One wave32 owns a 16-voxel tile (WMMA M dim) and all 64 output channels
// (4 N-tiles of 16). 8 waves / block share an 8 KB LDS stage of W[offset].
// ---------------------------------------------------------------------------
__global__ __launch_bounds__(256) void spconv_wmma_kernel(
    const float* __restrict__ feats,   // [N, 32]
    const float* __restrict__ Wg,      // [27, 32, 64]
    const int*   __restrict__ coors,   // [N, 4] = (b, z, y, x)
    float*       __restrict__ out,     // [B*64*64*64, 64]
    int N) {
    __shared__ float Ws[CIN * COUT];   // 8 KB: W tile for current offset

    const int tid   = threadIdx.x;
    const int wave  = tid >> 5;
    const int lane  = tid & 31;
    const int hl    = lane & 15;       // voxel slot within tile (M row)
    const int upper = lane >> 4;       // 0 -> K pair {0,1}; 1 -> K pair {2,3}

    const int  tile    = blockIdx.x * 8 + wave;
    const int  vm      = tile * 16 + hl;
    const bool inRange = (vm < N);
    const int  row     = inRange ? vm : (N - 1);

    // --- Gather A fragments once: lane holds feats[row][kk*4 + 2*upper + {0,1}]
    v2f af[8];
    const float* frow = feats + row * CIN;
    #pragma unroll
    for (int kk = 0; kk < 8; ++kk) {
        const float2 v = *reinterpret_cast<const float2*>(frow + kk * 4 + upper * 2);
        af[kk].x = v.x;
        af[kk].y = v.y;
    }

    // --- Per-dimension validity / output coordinate for each k in {0,1,2}
    const int bidx = coors[row * 4 + 0];
    const int pz   = coors[row * 4 + 1];
    const int py   = coors[row * 4 + 2];
    const int px   = coors[row * 4 + 3];

    int  oz[3], oy[3], ox[3];
    bool vz[3], vy[3], vx[3];
    #pragma unroll
    for (int k = 0; k < 3; ++k) {
        const int nz = pz + PADV - k;
        vz[k] = (nz >= 0) && ((nz & 1) == 0) && ((nz >> 1) < OUTD); oz[k] = nz >> 1;
        const int ny = py + PADV - k;
        vy[k] = (ny >= 0) && ((ny & 1) == 0) && ((ny >> 1) < OUTD); oy[k] = ny >> 1;
        const int nx = px + PADV - k;
        vx[k] = (nx >= 0) && ((nx & 1) == 0) && ((nx >> 1) < OUTD); ox[k] = nx >> 1;
    }

    for (int off = 0; off < 27; ++off) {
        const int kz = off / 9, ky = (off / 3) % 3, kx = off % 3;

        // Stage W[off] (32x64 f32 = 8 KB) into LDS, block-wide.
        // Barriers are executed by every wave on every iteration (uniform).
        __syncthreads();
        {
            const float4* src = reinterpret_cast<const float4*>(Wg + off * (CIN * COUT));
            float4*       dst = reinterpret_cast<float4*>(Ws);
            #pragma unroll
            for (int i = 0; i < (CIN * COUT / 4) / 256; ++i)
                dst[tid + i * 256] = src[tid + i * 256];
        }
        __syncthreads();

        const bool valid = inRange && vz[kz] && vy[ky] && vx[kx];
        if (!__any((int)valid)) continue;   // wave-uniform skip, EXEC stays full

        const int   flat   = valid ? (((bidx * OUTD + oz[kz]) * OUTD + oy[ky]) * OUTD + ox[kx]) : 0;
        const int   validI = valid ? 1 : 0;
        const float maskf  = valid ? 1.0f : 0.0f;

        // Row-mask the A fragments (zero contributions from invalid voxels).
        v2f am[8];
        #pragma unroll
        for (int kk = 0; kk < 8; ++kk) {
            am[kk].x = af[kk].x * maskf;
            am[kk].y = af[kk].y * maskf;
        }

        #pragma unroll
        for (int nt = 0; nt < 4; ++nt) {
            // 16x16 f32 accumulator tile: rows = voxels, cols = couts nt*16..+15
            v8f c = {};
            #pragma unroll
            for (int kk = 0; kk < 8; ++kk) {
                // B fragment 4x16: lane (hl) = column, upper half-wave = K rows {2,3}
                const int cin0 = kk * 4 + upper * 2;
                v2f b;
                b.x = Ws[(cin0    ) * COUT + nt * 16 + hl];
                b.y = Ws[(cin0 + 1) * COUT + nt * 16 + hl];
                c = __builtin_amdgcn_wmma_f32_16x16x4_f32(
                        /*neg_a=*/false, am[kk],
                        /*neg_b=*/false, b,
                        /*c_mod=*/(short)0, c,
                        /*reuse_a=*/false, /*reuse_b=*/false);
            }

            // Scatter: VGPR r holds (M = r + 8*upper, Ncol = nt*16 + hl).
            const int col = nt * 16 + hl;
            #pragma unroll
            for (int r = 0; r < 8; ++r) {
                const int m    = upper ? (r + 8) : r;
                const int fIdx = __shfl(flat,   m, 32);
                const int vIdx = __shfl(validI, m, 32);
                if (vIdx)
                    unsafeAtomicAdd(out + fIdx * COUT + col, c[r]);
            }
        }
    }
}

// ---------------------------------------------------------------------------
extern "C" void kernel_launch(void* const* d_in, const int* in_sizes, int n_in,
                              void* d_out, int out_size, void* d_ws, size_t ws_size,
                              hipStream_t stream) {
    const float* feats = (const float*)d_in[0];   // [N, 32] f32
    const float* Wg    = (const float*)d_in[1];   // [3,3,3,32,64] f32
    const int*   coors = (const int*)d_in[2];     // [N, 4] i32
    float*       out   = (float*)d_out;           // [2,64,64,64,64] f32

    const int N  = in_sizes[0] / CIN;             // 100000
    const int n4 = out_size / 4;                  // out_size divisible by 4

    spconv_zero_kernel<<<2048, 256, 0, stream>>>((float4*)out, n4);

    const int tiles  = (N + 15) / 16;
    const int blocks = (tiles + 7) / 8;           // 8 waves (tiles) per block
    spconv_wmma_kernel<<<blocks, 256, 0, stream>>>(feats, Wg, coors, out, N);
}